// GridMERAin16_33964601377389
// MI455X (gfx1250) — compile-verified
//
#include <hip/hip_runtime.h>
#include <hip/hip_bf16.h>

// ---------------------------------------------------------------------------
// GridMERA forward, MI455X (gfx1250, wave32).
//
// Roofline: dominant cost is the e1 contraction == GEMM
//   v[128,256] = P[128,65536] @ e1[65536,256]
// with P[B, a*4096+b*256+c*16+d] = p5[B,a]*p6[B,b]*p9[B,c]*p10[B,d] built on
// the fly.  e1 is 64 MB f32 -> HBM-bound (~2.8 us @ 23.3 TB/s), 4.3 GFLOPs.
// Therefore: f32 end-to-end, V_WMMA_F32_16X16X4_F32, LDS-staged e1 panels
// (one global read shared by 8 waves), split-K (64) + f32 global atomics.
// Panels are double-buffered via GLOBAL_LOAD_ASYNC_TO_LDS_B128 (ASYNCcnt)
// so the HBM fetch of panel p+1 overlaps the 16 WMMAs of panel p.
// All other tensors (<1 MB total traffic) handled by small scalar kernels.
// ---------------------------------------------------------------------------

typedef float v2f __attribute__((ext_vector_type(2)));
typedef float v8f __attribute__((ext_vector_type(8)));
typedef int   vi4 __attribute__((vector_size(4 * sizeof(int))));

#define BATCH   128
#define DD      16          // in_dims
#define BOND    4
#define OUTC    10
#define KTOT    65536       // 16^4
#define NTOT    256         // 4^4
#define KSPLIT  64
#define KCHUNK  (KTOT / KSPLIT)   // 1024
#define KPANEL  64
#define NPANEL  (KCHUNK / KPANEL) // 16

#if defined(__has_builtin)
#  if __has_builtin(__builtin_amdgcn_global_load_async_to_lds_b128)
#    define HAVE_ASYNC_LDS 1
#  endif
#endif
#ifndef HAVE_ASYNC_LDS
#  define HAVE_ASYNC_LDS 0
#endif

// ---- workspace layout (floats) -------------------------------------------
// xv   [128][16][16]        @ 0          (32768)
// vacc [128][256]           @ 32768      (32768)
// tbuf [4][128][16]         @ 65536      (8192)
// m2   [256][10]            @ 73728      (2560)
#define WS_XV    0
#define WS_VACC  32768
#define WS_TBUF  65536
#define WS_M2    73728

// ---------------------------------------------------------------------------
// Kernel 1: patch extraction + zero the split-K accumulator.
// xv[B, k=i*4+j, f=s1*4+s2] = inputs[B, i*4+s1, j*4+s2]
// ---------------------------------------------------------------------------
__global__ __launch_bounds__(256)
void k_prep(const float* __restrict__ in, float* __restrict__ xv,
            float* __restrict__ vacc) {
    int g = blockIdx.x * 256 + threadIdx.x;        // 0 .. 32767
    int B = g >> 8, rem = g & 255;
    int k = rem >> 4, f = rem & 15;
    int i = k >> 2, j = k & 3, s1 = f >> 2, s2 = f & 3;
    xv[g]   = in[B * 256 + (i * 4 + s1) * 16 + (j * 4 + s2)];
    vacc[g] = 0.0f;
}

// ---------------------------------------------------------------------------
// Kernel 2: the four layer-1 isometries -> tbuf[iso][B][m*4+o].
// t[B,m,o] = sum_{i,j,k} iso[i*si + j*sj + k*sk + m*sm + o] * pi[i]pj[j]pk[k]
// grid (32, 4), block 64  (4 batches x 16 (m,o) pairs per block)
// ---------------------------------------------------------------------------
__global__ __launch_bounds__(64)
void k_iso(const float* __restrict__ xv,
           const float* __restrict__ i0, const float* __restrict__ i1,
           const float* __restrict__ i2, const float* __restrict__ i3,
           float* __restrict__ tbuf) {
    const int iso = blockIdx.y;
    const int tid = threadIdx.x;
    const int bb = tid >> 4, mo = tid & 15, m = mo >> 2, o = mo & 3;
    const int B = blockIdx.x * 4 + bb;

    // per-iso leg strides (o stride is always 1) and patch triples
    const int SI[4] = {4096, 4096, 4096, 1024};
    const int SJ[4] = {256,  256,  64,   64  };
    const int SK[4] = {16,   4,    4,    4   };
    const int SM[4] = {4,    64,   1024, 16384};
    const int PK[4][3] = {{0,1,4},{2,3,7},{8,12,13},{11,14,15}};

    __shared__ float pr[4][3][16];                 // [batch-in-block][leg][feat]
    for (int e = tid; e < 192; e += 64) {
        int bx = e / 48, r = e % 48, pi = r >> 4, f = r & 15;
        pr[bx][pi][f] = xv[(blockIdx.x * 4 + bx) * 256 + PK[iso][pi] * 16 + f];
    }
    __syncthreads();

    const float* ip = (iso == 0) ? i0 : (iso == 1) ? i1 : (iso == 2) ? i2 : i3;
    const int si = SI[iso], sj = SJ[iso], sk = SK[iso], sm = SM[iso];

    float acc = 0.0f;
    for (int i = 0; i < 16; ++i)
        for (int j = 0; j < 16; ++j) {
            float pij = pr[bb][0][i] * pr[bb][1][j];
            int base = i * si + j * sj + m * sm + o;
            #pragma unroll 4
            for (int k = 0; k < 16; ++k)
                acc += ip[base + k * sk] * (pij * pr[bb][2][k]);
        }
    tbuf[iso * 2048 + B * 16 + mo] = acc;
}

// ---------------------------------------------------------------------------
// Kernel 3: batch-independent top contraction M2[abcd,C] = e2 @ iso2.
// ---------------------------------------------------------------------------
__global__ __launch_bounds__(256)
void k_m2(const float* __restrict__ e2, const float* __restrict__ iso2,
          float* __restrict__ m2) {
    int id = blockIdx.x * 256 + threadIdx.x;       // 0 .. 2559
    if (id >= 256 * OUTC) return;
    int row = id / OUTC, C = id % OUTC;
    float acc = 0.0f;
    for (int e = 0; e < 256; ++e)
        acc += e2[row * 256 + e] * iso2[e * OUTC + C];
    m2[id] = acc;
}

// ---------------------------------------------------------------------------
// Kernel 4: the big WMMA GEMM.  grid (KSPLIT=64, NT=16), block 256 (8 waves).
// wave w handles batch rows [16w, 16w+16); block handles N-tile blockIdx.y,
// K-chunk blockIdx.x.  e1 panels double-buffered in LDS via async-to-LDS
// DMA; A (the rank-4 product matrix P) is built on the fly from LDS-cached
// patch rows.
// ---------------------------------------------------------------------------
__global__ __launch_bounds__(256)
void k_gemm(const float* __restrict__ xv, const float* __restrict__ e1,
            float* __restrict__ vacc) {
    __shared__ float xvs[4][BATCH][17];            // padded: 17 coprime w/ 64 banks
    __shared__ float bpan[2][KPANEL][16];          // double-buffered e1 panel

    const int tid  = threadIdx.x;
    const int lane = tid & 31;
    const int wave = tid >> 5;
    const int half = lane >> 4;
    const int l15  = lane & 15;
    const int mb   = wave * 16 + l15;              // A-matrix row (batch)
    const int n0   = blockIdx.y * 16;
    const int kbase = blockIdx.x * KCHUNK;

    // each thread stages one 16B segment per panel: row kr, 16B segment q
    const int kr = tid >> 2;
    const int q  = tid & 3;

    // cache the four contracted patch rows (p5,p6,p9,p10) for all 128 batches
    const int PLEG[4] = {5, 6, 9, 10};
    for (int e = tid; e < 4 * BATCH * 16; e += 256) {
        int pi = e >> 11, r = e & 2047, bb = r >> 4, f = r & 15;
        xvs[pi][bb][f] = xv[bb * 256 + PLEG[pi] * 16 + f];
    }

    // panel stage: 64 rows x 64B of e1 -> bpan[buf]
    auto issue_panel = [&](int p, int buf) {
        const float* gsrc = e1 + (size_t)(kbase + p * KPANEL + kr) * NTOT + n0 + q * 4;
        float* ldst = &bpan[buf][kr][q * 4];
#if HAVE_ASYNC_LDS
        __builtin_amdgcn_global_load_async_to_lds_b128(
            (__attribute__((address_space(1))) vi4*)(float*)gsrc,
            (__attribute__((address_space(3))) vi4*)ldst,
            /*offset=*/0, /*cpol=*/0);
#else
        *(float4*)ldst = *(const float4*)gsrc;
#endif
    };

    issue_panel(0, 0);                             // prologue: panel 0 in flight

    v8f acc = {};
    for (int p = 0; p < NPANEL; ++p) {
        if (p + 1 < NPANEL) {
            issue_panel(p + 1, (p + 1) & 1);       // overlap next fetch w/ compute
#if HAVE_ASYNC_LDS
            asm volatile("s_wait_asynccnt 0x1" ::: "memory");
#endif
        } else {
#if HAVE_ASYNC_LDS
            asm volatile("s_wait_asynccnt 0x0" ::: "memory");
#endif
        }
        __syncthreads();                           // panel p visible to all waves

        const int kp = kbase + p * KPANEL;
        const float (* __restrict__ bp)[16] = bpan[p & 1];

        // a,b legs of K are constant inside an aligned 64-row panel
        const float pab = xvs[0][mb][(kp >> 12) & 15] * xvs[1][mb][(kp >> 8) & 15];

        #pragma unroll
        for (int kk = 0; kk < KPANEL; kk += 4) {
            const int ka  = kk + half * 2;         // K slot for this half-wave
            const int kg  = kp + ka;
            v2f A, Bv;
            A.x  = pab * xvs[2][mb][(kg >> 4) & 15] * xvs[3][mb][kg & 15];
            A.y  = pab * xvs[2][mb][((kg + 1) >> 4) & 15] * xvs[3][mb][(kg + 1) & 15];
            Bv.x = bp[ka][l15];
            Bv.y = bp[ka + 1][l15];
            acc = __builtin_amdgcn_wmma_f32_16x16x4_f32(
                /*neg_a=*/false, A, /*neg_b=*/false, Bv,
                /*c_mod=*/(short)0, acc, /*reuse_a=*/false, /*reuse_b=*/false);
        }
        __syncthreads();                           // all reads done before reuse
    }

    // C/D layout: reg r, lanes 0-15 -> M=r, lanes 16-31 -> M=r+8
    #pragma unroll
    for (int r = 0; r < 8; ++r) {
        int row = wave * 16 + r + half * 8;
        atomicAdd(&vacc[row * NTOT + n0 + l15], acc[r]);
    }
}

// ---------------------------------------------------------------------------
// Kernel 5: fuse v with t0..t3 (four bond rotations), then project with M2.
// One block per batch, 256 threads (one per 4^4 multi-index).
// Each stage: next[u,v,w,z'] = sum_s prev[s*64 + (tid>>2)] * t[s*4 + (tid&3)]
// ---------------------------------------------------------------------------
__global__ __launch_bounds__(256)
void k_fuse(const float* __restrict__ vacc, const float* __restrict__ tbuf,
            const float* __restrict__ m2, const float* __restrict__ bias,
            float* __restrict__ out) {
    __shared__ float buf0[256], buf1[256];
    __shared__ float tsh[4][16];
    const int B = blockIdx.x, tid = threadIdx.x;

    buf0[tid] = vacc[B * 256 + tid];
    if (tid < 64) tsh[tid >> 4][tid & 15] = tbuf[(tid >> 4) * 2048 + B * 16 + (tid & 15)];
    __syncthreads();

    const int hi = tid >> 2, lo = tid & 3;
    #pragma unroll
    for (int st = 0; st < 4; ++st) {
        const float* src = (st & 1) ? buf1 : buf0;
        float*       dst = (st & 1) ? buf0 : buf1;
        float s = 0.0f;
        #pragma unroll
        for (int e = 0; e < BOND; ++e)
            s += src[e * 64 + hi] * tsh[st][e * 4 + lo];
        dst[tid] = s;
        __syncthreads();
    }
    // after 4 stages result (w[abcd]) is in buf0
    if (tid < OUTC) {
        float s = bias[tid];
        for (int qd = 0; qd < 256; ++qd)
            s += buf0[qd] * m2[qd * OUTC + tid];
        out[B * OUTC + tid] = s;
    }
}

// ---------------------------------------------------------------------------
extern "C" void kernel_launch(void* const* d_in, const int* in_sizes, int n_in,
                              void* d_out, int out_size, void* d_ws, size_t ws_size,
                              hipStream_t stream) {
    const float* in   = (const float*)d_in[0];
    const float* e1   = (const float*)d_in[1];
    const float* e2   = (const float*)d_in[2];
    const float* i0   = (const float*)d_in[3];
    const float* i1   = (const float*)d_in[4];
    const float* i2   = (const float*)d_in[5];
    const float* i3   = (const float*)d_in[6];
    const float* iso2 = (const float*)d_in[7];
    const float* bias = (const float*)d_in[8];
    float* out = (float*)d_out;

    float* ws   = (float*)d_ws;
    float* xv   = ws + WS_XV;
    float* vacc = ws + WS_VACC;
    float* tbuf = ws + WS_TBUF;
    float* m2   = ws + WS_M2;

    k_prep<<<dim3(BATCH), dim3(256), 0, stream>>>(in, xv, vacc);
    k_iso <<<dim3(32, 4), dim3(64),  0, stream>>>(xv, i0, i1, i2, i3, tbuf);
    k_m2  <<<dim3(10),    dim3(256), 0, stream>>>(e2, iso2, m2);
    k_gemm<<<dim3(KSPLIT, 16), dim3(256), 0, stream>>>(xv, e1, vacc);
    k_fuse<<<dim3(BATCH), dim3(256), 0, stream>>>(vacc, tbuf, m2, bias, out);
}